// GCN_25134148616642
// MI455X (gfx1250) — compile-verified
//
#include <hip/hip_runtime.h>
#include <math.h>

typedef float v2f __attribute__((ext_vector_type(2)));
typedef float v8f __attribute__((ext_vector_type(8)));

#define NNODES 100000
#define NEDGES 3200000
#define NTILES (NNODES / 16)   // 6250, exact

// -------------------------------------------------------------------------
// Degree kernels: deg[i] = 1 + sum_{e: dst=i} 1 ; then dinv = rsqrt(deg)
// -------------------------------------------------------------------------
__global__ void deg_init(float* __restrict__ dinv, int n) {
  int i = blockIdx.x * blockDim.x + threadIdx.x;
  if (i < n) dinv[i] = 1.0f;
}

__global__ void deg_count(const int* __restrict__ dst, float* __restrict__ dinv, int e) {
  int i = blockIdx.x * blockDim.x + threadIdx.x;
  if (i < e) atomicAdd(&dinv[dst[i]], 1.0f);
}

__global__ void deg_rsqrt(float* __restrict__ dinv, int n) {
  int i = blockIdx.x * blockDim.x + threadIdx.x;
  if (i < n) dinv[i] = rsqrtf(dinv[i]);
}

// -------------------------------------------------------------------------
// WMMA f32 GEMM: Y[16*numTiles x NCOLS] = X[.. x K] * W[K x NCOLS]
// One wave32 per 16-row tile. Uses V_WMMA_F32_16X16X4_F32 (exact f32 path).
// A fragment: lane L (m=L&15, grp=L>>4) holds X[row][k0+2*grp+{0,1}]
// B fragment: vgpr v, lane L holds W[k0+2*grp+v][L&15 (+ 16*j)]
// D fragment: vgpr r, lane L holds Y[tile*16 + r + 8*grp][L&15 (+ 16*j)]
// -------------------------------------------------------------------------
template<int K, int NCOLS>
__global__ void gemm_wmma(const float* __restrict__ X, const float* __restrict__ W,
                          float* __restrict__ Y, int numTiles) {
  __shared__ float sW[K * NCOLS];
  for (int i = threadIdx.x; i < K * NCOLS; i += blockDim.x) sW[i] = W[i];
  __syncthreads();

  const int wavesPerBlock = blockDim.x >> 5;
  const int tile = blockIdx.x * wavesPerBlock + (threadIdx.x >> 5);
  if (tile >= numTiles) return;

  const int lane = threadIdx.x & 31;
  const int grp  = lane >> 4;    // 0: K pair {0,1}; 1: K pair {2,3}
  const int m    = lane & 15;

  v8f acc[NCOLS / 16];
#pragma unroll
  for (int j = 0; j < NCOLS / 16; ++j) acc[j] = (v8f)0.0f;

  const float* xrow = X + (size_t)(tile * 16 + m) * K;

#pragma unroll 4
  for (int k0 = 0; k0 < K; k0 += 4) {
    v2f a;
    a.x = xrow[k0 + 2 * grp + 0];
    a.y = xrow[k0 + 2 * grp + 1];
#pragma unroll
    for (int j = 0; j < NCOLS / 16; ++j) {
      v2f b;
      const int col = j * 16 + m;
      b.x = sW[(k0 + 2 * grp + 0) * NCOLS + col];
      b.y = sW[(k0 + 2 * grp + 1) * NCOLS + col];
      acc[j] = __builtin_amdgcn_wmma_f32_16x16x4_f32(
          /*neg_a=*/false, a, /*neg_b=*/false, b,
          /*c_mod=*/(short)0, acc[j], /*reuse_a=*/false, /*reuse_b=*/false);
    }
  }

#pragma unroll
  for (int j = 0; j < NCOLS / 16; ++j) {
#pragma unroll
    for (int r = 0; r < 8; ++r) {
      const int mo = r + 8 * grp;
      Y[(size_t)(tile * 16 + mo) * NCOLS + j * 16 + m] = acc[j][r];
    }
  }
}

// -------------------------------------------------------------------------
// Aggregation: acc[i] = dinv[i]^2 * hlin[i]           (self-loop init)
//              acc[d] += dinv[s]*dinv[d]*hlin[s]      (edge scatter-atomic)
//              h[i]    = elu(acc[i] + bias)           (finish, in place)
// D = 1<<LOGD features, one thread per (row, feature) / (edge, feature).
// -------------------------------------------------------------------------
template<int LOGD>
__global__ void agg_self(const float* __restrict__ dinv, const float* __restrict__ hlin,
                         float* __restrict__ acc, int total) {
  int t = blockIdx.x * blockDim.x + threadIdx.x;
  if (t < total) {
    int node = t >> LOGD;
    float w = dinv[node];
    acc[t] = w * w * hlin[t];
  }
}

template<int LOGD>
__global__ void agg_edges(const int* __restrict__ src, const int* __restrict__ dst,
                          const float* __restrict__ dinv, const float* __restrict__ hlin,
                          float* __restrict__ acc, int total) {
  const int D = 1 << LOGD;
  int t = blockIdx.x * blockDim.x + threadIdx.x;
  if (t < total) {
    int e = t >> LOGD;
    int j = t & (D - 1);
    int s = src[e];
    int d = dst[e];
    float v = dinv[s] * dinv[d] * hlin[(size_t)s * D + j];
    atomicAdd(&acc[(size_t)d * D + j], v);
  }
}

template<int LOGD>
__global__ void agg_finish(float* __restrict__ acc, const float* __restrict__ bias, int total) {
  const int D = 1 << LOGD;
  int t = blockIdx.x * blockDim.x + threadIdx.x;
  if (t < total) {
    float v = acc[t] + bias[t & (D - 1)];
    acc[t] = (v > 0.0f) ? v : (expf(v) - 1.0f);   // ELU, alpha = 1
  }
}

// -------------------------------------------------------------------------
// log_softmax over 32 classes, one thread per node (rows are L2-resident)
// -------------------------------------------------------------------------
__global__ void log_softmax32(const float* __restrict__ h, float* __restrict__ out, int n) {
  int i = blockIdx.x * blockDim.x + threadIdx.x;
  if (i >= n) return;
  const float* row = h + (size_t)i * 32;
  float m = row[0];
#pragma unroll
  for (int j = 1; j < 32; ++j) m = fmaxf(m, row[j]);
  float s = 0.0f;
#pragma unroll
  for (int j = 0; j < 32; ++j) s += expf(row[j] - m);
  float ls = logf(s);
  float* orow = out + (size_t)i * 32;
#pragma unroll
  for (int j = 0; j < 32; ++j) orow[j] = row[j] - m - ls;
}

// -------------------------------------------------------------------------
extern "C" void kernel_launch(void* const* d_in, const int* in_sizes, int n_in,
                              void* d_out, int out_size, void* d_ws, size_t ws_size,
                              hipStream_t stream) {
  const float* x  = (const float*)d_in[0];
  const int*   ei = (const int*)d_in[1];
  const float* W1 = (const float*)d_in[2];
  const float* b1 = (const float*)d_in[3];
  const float* W2 = (const float*)d_in[4];
  const float* b2 = (const float*)d_in[5];
  const float* W3 = (const float*)d_in[6];
  const float* b3 = (const float*)d_in[7];
  float* out = (float*)d_out;

  const int* src = ei;            // edge_index[0]
  const int* dst = ei + NEDGES;   // edge_index[1]

  float* ws   = (float*)d_ws;
  float* dinv = ws;                          // N floats
  float* bufA = dinv + NNODES;               // N*32 floats
  float* bufB = bufA + (size_t)NNODES * 32;  // N*32 floats

  const int TPB = 256;
  auto blocks = [](long long total, int tpb) { return (int)((total + tpb - 1) / tpb); };

  // --- degrees ---
  deg_init <<<blocks(NNODES, TPB), TPB, 0, stream>>>(dinv, NNODES);
  deg_count<<<blocks(NEDGES, TPB), TPB, 0, stream>>>(dst, dinv, NEDGES);
  deg_rsqrt<<<blocks(NNODES, TPB), TPB, 0, stream>>>(dinv, NNODES);

  const int gemmBlocks = blocks(NTILES, TPB / 32);   // 8 waves (tiles) per block

  // --- layer 1: x[N,256] @ W1[256,16] -> bufA ; aggregate -> bufB ---
  gemm_wmma<256, 16><<<gemmBlocks, TPB, 0, stream>>>(x, W1, bufA, NTILES);
  agg_self <4><<<blocks((long long)NNODES * 16, TPB), TPB, 0, stream>>>(dinv, bufA, bufB, NNODES * 16);
  agg_edges<4><<<blocks((long long)NEDGES * 16, TPB), TPB, 0, stream>>>(src, dst, dinv, bufA, bufB, NEDGES * 16);
  agg_finish<4><<<blocks((long long)NNODES * 16, TPB), TPB, 0, stream>>>(bufB, b1, NNODES * 16);

  // --- layer 2: bufB[N,16] @ W2[16,16] -> bufA ; aggregate -> bufB ---
  gemm_wmma<16, 16><<<gemmBlocks, TPB, 0, stream>>>(bufB, W2, bufA, NTILES);
  agg_self <4><<<blocks((long long)NNODES * 16, TPB), TPB, 0, stream>>>(dinv, bufA, bufB, NNODES * 16);
  agg_edges<4><<<blocks((long long)NEDGES * 16, TPB), TPB, 0, stream>>>(src, dst, dinv, bufA, bufB, NEDGES * 16);
  agg_finish<4><<<blocks((long long)NNODES * 16, TPB), TPB, 0, stream>>>(bufB, b2, NNODES * 16);

  // --- layer 3: bufB[N,16] @ W3[16,32] -> bufA ; aggregate -> bufB ---
  gemm_wmma<16, 32><<<gemmBlocks, TPB, 0, stream>>>(bufB, W3, bufA, NTILES);
  agg_self <5><<<blocks((long long)NNODES * 32, TPB), TPB, 0, stream>>>(dinv, bufA, bufB, NNODES * 32);
  agg_edges<5><<<blocks((long long)NEDGES * 32, TPB), TPB, 0, stream>>>(src, dst, dinv, bufA, bufB, NEDGES * 32);
  agg_finish<5><<<blocks((long long)NNODES * 32, TPB), TPB, 0, stream>>>(bufB, b3, NNODES * 32);

  // --- classifier head ---
  log_softmax32<<<blocks(NNODES, TPB), TPB, 0, stream>>>(bufB, out, NNODES);
}